// QuantumLayerVQC_1949915153145
// MI455X (gfx1250) — compile-verified
//
#include <hip/hip_runtime.h>

typedef __attribute__((ext_vector_type(2))) float v2f;
typedef __attribute__((ext_vector_type(8))) float v8f;

// ---------------------------------------------------------------------------
// Kernel 1: build the 8x8 real symmetric matrix A = Re(U^H Z0 U) from the 18
// circuit weights. U = (layers of RY/RZ + CNOT01 + CNOT12) applied to the 8
// basis states. Trivial work: 1 block of 64 threads.
// ---------------------------------------------------------------------------
__global__ void vqc_build_A(const float* __restrict__ qw, float* __restrict__ Aout) {
  __shared__ float Ur[8][8]; // Ur[k][col] = Re U[k][col]
  __shared__ float Ui[8][8];
  const int t = threadIdx.x;
  if (t < 8) {
    float sr[8], si[8];
#pragma unroll
    for (int i = 0; i < 8; ++i) { sr[i] = 0.f; si[i] = 0.f; }
    sr[t] = 1.f; // basis state |t>
    for (int l = 0; l < 3; ++l) {
      for (int q = 0; q < 3; ++q) {
        const int bit = 4 >> q; // q0->4, q1->2, q2->1
        float sy, cy, sz, cz;
        __sincosf(0.5f * qw[(l * 3 + q) * 2 + 0], &sy, &cy);
        __sincosf(0.5f * qw[(l * 3 + q) * 2 + 1], &sz, &cz);
        // RY(theta) = [[c,-s],[s,c]] on qubit q (real, applied to re & im)
#pragma unroll
        for (int i = 0; i < 8; ++i) {
          if (!(i & bit)) {
            const int j = i | bit;
            float r0 = sr[i], i0 = si[i], r1 = sr[j], i1 = si[j];
            sr[i] = cy * r0 - sy * r1;  si[i] = cy * i0 - sy * i1;
            sr[j] = sy * r0 + cy * r1;  si[j] = sy * i0 + cy * i1;
          }
        }
        // RZ(theta) = diag(e^{-ih}, e^{+ih})
#pragma unroll
        for (int i = 0; i < 8; ++i) {
          const float sg = (i & bit) ? sz : -sz;
          const float r = sr[i], im = si[i];
          sr[i] = cz * r - sg * im;
          si[i] = cz * im + sg * r;
        }
      }
      // CNOT01: swap amps 4<->6, 5<->7 ; CNOT12: swap 2<->3, 6<->7
      float tr, ti;
      tr = sr[4]; sr[4] = sr[6]; sr[6] = tr;  ti = si[4]; si[4] = si[6]; si[6] = ti;
      tr = sr[5]; sr[5] = sr[7]; sr[7] = tr;  ti = si[5]; si[5] = si[7]; si[7] = ti;
      tr = sr[2]; sr[2] = sr[3]; sr[3] = tr;  ti = si[2]; si[2] = si[3]; si[3] = ti;
      tr = sr[6]; sr[6] = sr[7]; sr[7] = tr;  ti = si[6]; si[6] = si[7]; si[7] = ti;
    }
#pragma unroll
    for (int k = 0; k < 8; ++k) { Ur[k][t] = sr[k]; Ui[k][t] = si[k]; }
  }
  __syncthreads();
  if (t < 64) {
    const int i = t >> 3, j = t & 7;
    float acc = 0.f;
#pragma unroll
    for (int k = 0; k < 8; ++k) {
      const float d = (k < 4) ? 1.f : -1.f; // Z on qubit 0 (msb of index)
      acc += d * (Ur[k][i] * Ur[k][j] + Ui[k][i] * Ui[k][j]);
    }
    Aout[t] = acc; // row-major A[i][j]
  }
}

// ---------------------------------------------------------------------------
// Kernel 2: streaming kernel. Per wave: 16 samples (sample = lane & 15, both
// lane halves compute the same sample's encoding vector e so the WMMA B
// operand is lane-local). y = Apad(16x8) x E(8x16) via two
// V_WMMA_F32_16X16X4_F32, then z = dot(e, y) per lane.
// ---------------------------------------------------------------------------
__global__ __launch_bounds__(256)
void vqc_main(const float* __restrict__ in, const float* __restrict__ Ag,
              float* __restrict__ out, int B) {
  __shared__ float As[64];
  const int tid = threadIdx.x;
  if (tid < 64) As[tid] = Ag[tid];
  __syncthreads();

  const int lane = tid & 31;
  const int wave = tid >> 5;
  const int r    = lane & 15;
  const bool hi  = lane >= 16;

  const int base   = (blockIdx.x * 8 + wave) * 16;
  int sample       = base + r;
  const int samp_c = (sample < B) ? sample : (B - 1); // clamp, keep EXEC full

  // ---- load first 16B of the 32B input row; use x,y,z -------------------
  const float4 v = *(const float4*)(in + (size_t)samp_c * 8);

  float c0, s0, c1, s1, c2, s2;
  __sincosf(0.5f * v.x, &s0, &c0);
  __sincosf(0.5f * v.y, &s1, &c1);
  __sincosf(0.5f * v.z, &s2, &c2);

  // encoding state e[q0*4 + q1*2 + q2] = f0[q0]*f1[q1]*f2[q2]
  float e[8];
  const float cc = c1 * c2, cs = c1 * s2, sc = s1 * c2, ss = s1 * s2;
  e[0] = c0 * cc; e[1] = c0 * cs; e[2] = c0 * sc; e[3] = c0 * ss;
  e[4] = s0 * cc; e[5] = s0 * cs; e[6] = s0 * sc; e[7] = s0 * ss;

  // ---- WMMA A operand: Apad rows (M = r, zero for r>=8), K via lane half --
  // 16x4 f32 A layout: lanes 0-15 -> K=0,1 in v0,v1 ; lanes 16-31 -> K=2,3
  v2f a01 = {0.f, 0.f}, a23 = {0.f, 0.f};
  if (r < 8) {
    const float* Ar = &As[r * 8];
    a01.x = Ar[hi ? 2 : 0];  a01.y = Ar[hi ? 3 : 1];
    a23.x = Ar[hi ? 6 : 4];  a23.y = Ar[hi ? 7 : 5];
  }

  // ---- WMMA B operand: E (8x16, K x N), N = lane&15, lane-local ----------
  // 4x16 f32 B layout mirrors A: lanes 0-15 -> K=0,1 ; lanes 16-31 -> K=2,3
  v2f b01, b23;
  b01.x = hi ? e[2] : e[0];  b01.y = hi ? e[3] : e[1];
  b23.x = hi ? e[6] : e[4];  b23.y = hi ? e[7] : e[5];

  // ---- y = Apad x E, accumulated over K=0..3 and K=4..7 ------------------
  v8f acc = {};
  acc = __builtin_amdgcn_wmma_f32_16x16x4_f32(false, a01, false, b01,
                                              (short)0, acc, false, false);
  acc = __builtin_amdgcn_wmma_f32_16x16x4_f32(false, a23, false, b23,
                                              (short)0, acc, false, false);

  // D layout: lane n<16, VGPR m -> (M=m, N=n) = y_m of sample n.
  // Rows 8..15 of Apad are zero, so lanes 16-31 hold zeros; lanes 0-15 finish.
  const float z = e[0] * acc[0] + e[1] * acc[1] + e[2] * acc[2] + e[3] * acc[3] +
                  e[4] * acc[4] + e[5] * acc[5] + e[6] * acc[6] + e[7] * acc[7];

  if (!hi && sample < B) out[sample] = z;
}

// ---------------------------------------------------------------------------
extern "C" void kernel_launch(void* const* d_in, const int* in_sizes, int n_in,
                              void* d_out, int out_size, void* d_ws, size_t ws_size,
                              hipStream_t stream) {
  const float* inputs = (const float*)d_in[0];   // (B, 8) f32
  const float* qw     = (const float*)d_in[1];   // (3, 3, 2) f32
  float* out          = (float*)d_out;           // (B, 1) f32
  float* A            = (float*)d_ws;            // 64 f32 scratch

  const int B = in_sizes[0] / 8;

  vqc_build_A<<<1, 64, 0, stream>>>(qw, A);

  const int samples_per_block = 128; // 8 waves x 16 samples
  const int blocks = (B + samples_per_block - 1) / samples_per_block;
  vqc_main<<<blocks, 256, 0, stream>>>(inputs, A, out, B);
}